// TopKDiceLoss_3556232921393
// MI455X (gfx1250) — compile-verified
//
#include <hip/hip_runtime.h>

typedef unsigned int u32;
typedef __attribute__((ext_vector_type(2))) float v2f;
typedef __attribute__((ext_vector_type(8))) float v8f;

#define NBATCH 32
#define NPIX   (512*512)
#define NPIX2  (2*NPIX)
#define BINS_A 2048
#define BINS_C 1024
#define CPB    32                     // blocks per batch in pass kernels
#define TPB    256                    // 8 wave32 per block
#define EPB    (NPIX / CPB)           // 8192 elements per block
#define PITER  (EPB / (TPB*4))        // 8 float4 iterations per thread
#define NWAVES (TPB/32)

// workspace layout (u32 units)
#define OFF_H1  0
#define OFF_H2  (NBATCH*BINS_A)
#define OFF_H3  (2*NBATCH*BINS_A)
#define OFF_ACC (2*NBATCH*BINS_A + NBATCH*BINS_C)
#define ACC_ST  16
#define WS_U32  (OFF_ACC + NBATCH*ACC_ST)
#define OFF_TP  WS_U32                 // optional tp-bits cache (NBATCH*NPIX u32)
#define WS_U32_CACHED (OFF_TP + NBATCH*NPIX)
// acc slots: 0 sumP2(f) 1 n_fg(f) 2 d1 3 r1 4 d2 5 r2 6 thr_bits 7 inter(f) 8 p2_ign(f) 9 cnt_keep(f)

#define SMOOTHF 1e-5f
#define FG_SENTINEL 0xFFFFFFFFu        // digit0 = 2047 > 511 max real digit (tp < 2.0)

__device__ __forceinline__ float sig_p(float l0, float l1) {
  // softmax over 2 channels, channel 1 == sigmoid(l1 - l0)
  return 1.0f / (1.0f + expf(l0 - l1));
}

// Wave-wide sum of one float per lane via V_WMMA_F32_16X16X4_F32:
// A = ones(16x4), B holds per-lane value in VGPR0 (zero in VGPR1).
// D[m][n] = colsum[n]; VGPR0 of D carries rows M=0 and M=8, so
// sum of c[0] over all 32 lanes == 2 * total.
__device__ __forceinline__ float wave_sum(float v) {
#if __has_builtin(__builtin_amdgcn_wmma_f32_16x16x4_f32)
  v2f a;  a[0] = 1.0f; a[1] = 1.0f;
  v2f bm; bm[0] = v;   bm[1] = 0.0f;
  v8f c = {};
  c = __builtin_amdgcn_wmma_f32_16x16x4_f32(false, a, false, bm,
                                            (short)0, c, false, false);
  float s = c[0];
  #pragma unroll
  for (int off = 16; off >= 1; off >>= 1) s += __shfl_xor(s, off, 32);
  return 0.5f * s;
#else
  #pragma unroll
  for (int off = 16; off >= 1; off >>= 1) v += __shfl_xor(v, off, 32);
  return v;
#endif
}

__device__ __forceinline__ void block_atomic_add(float v, float* dst, float* red) {
  int lane = threadIdx.x & 31;
  int wv   = threadIdx.x >> 5;
  float t = wave_sum(v);              // requires full EXEC: called with all threads active
  if (lane == 0) red[wv] = t;
  __syncthreads();
  if (threadIdx.x == 0) {
    float s = 0.0f;
    #pragma unroll
    for (int i = 0; i < NWAVES; ++i) s += red[i];
    atomicAdd(dst, s);
  }
  __syncthreads();
}

// ---------------- zero workspace (hists + accums only) ----------------
__global__ void k_zero(u32* __restrict__ ws) {
  u32 i = blockIdx.x * blockDim.x + threadIdx.x;
  if (i < WS_U32) ws[i] = 0u;
}

// =====================================================================
// Pass 1 (shared by both paths): digit0 histogram + sum(p^2) + n_fg.
// Cached variant also writes tp bit patterns (sentinel for non-fg).
// =====================================================================
template <bool CACHE>
__device__ __forceinline__ u32 acc1(float l0, float l1, float t, float e,
                                    float& p2, float& fg, u32* hist) {
  float p = sig_p(l0, l1);
  p2 += p * p;
  u32 bits = FG_SENTINEL;
  if (t == 1.0f) {
    float tp = p * (1.0f + e);
    bits = __float_as_uint(tp);
    atomicAdd(&hist[bits >> 21], 1u);
    fg += 1.0f;
  }
  return bits;
}

template <bool CACHE>
__global__ __launch_bounds__(TPB) void k_pass1(const float* __restrict__ logits,
                                               const float* __restrict__ target,
                                               const float* __restrict__ eps,
                                               u32* __restrict__ ws) {
  __shared__ u32 hist[BINS_A];
  __shared__ float red[NWAVES];
  int b = blockIdx.x / CPB, chunk = blockIdx.x % CPB;
  for (int i = threadIdx.x; i < BINS_A; i += TPB) hist[i] = 0u;
  __syncthreads();

  const float* l0p = logits + (size_t)b * NPIX2;
  const float* l1p = l0p + NPIX;
  const float* tpp = target + (size_t)b * NPIX;
  const float* epp = eps    + (size_t)b * NPIX;
  u32* tpb = ws + OFF_TP + (size_t)b * NPIX;
  int base = chunk * EPB;
  float p2 = 0.0f, fg = 0.0f;
  #pragma unroll
  for (int it = 0; it < PITER; ++it) {
    int i = base + it * (TPB * 4) + threadIdx.x * 4;
    float4 a0 = *(const float4*)(l0p + i);
    float4 a1 = *(const float4*)(l1p + i);
    float4 tt = *(const float4*)(tpp + i);
    float4 ee = *(const float4*)(epp + i);
    uint4 q;
    q.x = acc1<CACHE>(a0.x, a1.x, tt.x, ee.x, p2, fg, hist);
    q.y = acc1<CACHE>(a0.y, a1.y, tt.y, ee.y, p2, fg, hist);
    q.z = acc1<CACHE>(a0.z, a1.z, tt.z, ee.z, p2, fg, hist);
    q.w = acc1<CACHE>(a0.w, a1.w, tt.w, ee.w, p2, fg, hist);
    if (CACHE) *(uint4*)(tpb + i) = q;
  }
  __syncthreads();
  u32* gh = ws + OFF_H1 + (size_t)b * BINS_A;
  for (int i = threadIdx.x; i < BINS_A; i += TPB) {
    u32 c = hist[i];
    if (c) atomicAdd(&gh[i], c);
  }
  float* accf = (float*)(ws + OFF_ACC + (size_t)b * ACC_ST);
  block_atomic_add(p2, &accf[0], red);
  block_atomic_add(fg, &accf[1], red);
}

// =====================================================================
// CACHED path: passes 2/3/final read tp bits (4 B/pixel) instead of
// re-reading 16 B/pixel of inputs and redoing the sigmoid.
// =====================================================================
__global__ __launch_bounds__(TPB) void k_pass2c(u32* __restrict__ ws) {
  __shared__ u32 hist[BINS_A];
  int b = blockIdx.x / CPB, chunk = blockIdx.x % CPB;
  u32* acc = ws + OFF_ACC + (size_t)b * ACC_ST;
  u32 d1 = acc[2];
  if (d1 == FG_SENTINEL) return;
  for (int i = threadIdx.x; i < BINS_A; i += TPB) hist[i] = 0u;
  __syncthreads();
  const u32* tpb = ws + OFF_TP + (size_t)b * NPIX;
  int base = chunk * EPB;
  #pragma unroll
  for (int it = 0; it < PITER; ++it) {
    int i = base + it * (TPB * 4) + threadIdx.x * 4;
    uint4 q = *(const uint4*)(tpb + i);
    if ((q.x >> 21) == d1) atomicAdd(&hist[(q.x >> 10) & 0x7FFu], 1u);
    if ((q.y >> 21) == d1) atomicAdd(&hist[(q.y >> 10) & 0x7FFu], 1u);
    if ((q.z >> 21) == d1) atomicAdd(&hist[(q.z >> 10) & 0x7FFu], 1u);
    if ((q.w >> 21) == d1) atomicAdd(&hist[(q.w >> 10) & 0x7FFu], 1u);
  }
  __syncthreads();
  u32* gh = ws + OFF_H2 + (size_t)b * BINS_A;
  for (int i = threadIdx.x; i < BINS_A; i += TPB) {
    u32 c = hist[i];
    if (c) atomicAdd(&gh[i], c);
  }
}

__global__ __launch_bounds__(TPB) void k_pass3c(u32* __restrict__ ws) {
  __shared__ u32 hist[BINS_C];
  int b = blockIdx.x / CPB, chunk = blockIdx.x % CPB;
  u32* acc = ws + OFF_ACC + (size_t)b * ACC_ST;
  u32 d1 = acc[2];
  if (d1 == FG_SENTINEL) return;
  u32 key = (d1 << 11) | acc[4];
  for (int i = threadIdx.x; i < BINS_C; i += TPB) hist[i] = 0u;
  __syncthreads();
  const u32* tpb = ws + OFF_TP + (size_t)b * NPIX;
  int base = chunk * EPB;
  #pragma unroll
  for (int it = 0; it < PITER; ++it) {
    int i = base + it * (TPB * 4) + threadIdx.x * 4;
    uint4 q = *(const uint4*)(tpb + i);
    if ((q.x >> 10) == key) atomicAdd(&hist[q.x & 0x3FFu], 1u);
    if ((q.y >> 10) == key) atomicAdd(&hist[q.y & 0x3FFu], 1u);
    if ((q.z >> 10) == key) atomicAdd(&hist[q.z & 0x3FFu], 1u);
    if ((q.w >> 10) == key) atomicAdd(&hist[q.w & 0x3FFu], 1u);
  }
  __syncthreads();
  u32* gh = ws + OFF_H3 + (size_t)b * BINS_C;
  for (int i = threadIdx.x; i < BINS_C; i += TPB) {
    u32 c = hist[i];
    if (c) atomicAdd(&gh[i], c);
  }
}

__device__ __forceinline__ void acc4c(u32 bits, float e, float thr,
                                      float& inter, float& cnt, float& p2ig) {
  if (bits != FG_SENTINEL) {
    float tp = __uint_as_float(bits);
    float p  = tp / (1.0f + e);                 // recover probs for fg pixel
    if (tp <= thr) { inter += p; cnt += 1.0f; } // kept fg
    else           { p2ig += p * p; }           // ignored fg
  }
}

__global__ __launch_bounds__(TPB) void k_finalc(const float* __restrict__ eps,
                                                u32* __restrict__ ws) {
  __shared__ float red[NWAVES];
  int b = blockIdx.x / CPB, chunk = blockIdx.x % CPB;
  u32* acc = ws + OFF_ACC + (size_t)b * ACC_ST;
  float thr = __uint_as_float(acc[6]);
  const u32*   tpb = ws  + OFF_TP + (size_t)b * NPIX;
  const float* epp = eps + (size_t)b * NPIX;
  int base = chunk * EPB;
  float inter = 0.0f, cnt = 0.0f, p2ig = 0.0f;
  #pragma unroll
  for (int it = 0; it < PITER; ++it) {
    int i = base + it * (TPB * 4) + threadIdx.x * 4;
    uint4  q  = *(const uint4*)(tpb + i);
    float4 ee = *(const float4*)(epp + i);
    acc4c(q.x, ee.x, thr, inter, cnt, p2ig);
    acc4c(q.y, ee.y, thr, inter, cnt, p2ig);
    acc4c(q.z, ee.z, thr, inter, cnt, p2ig);
    acc4c(q.w, ee.w, thr, inter, cnt, p2ig);
  }
  float* accf = (float*)acc;
  block_atomic_add(inter, &accf[7], red);
  block_atomic_add(p2ig,  &accf[8], red);
  block_atomic_add(cnt,   &accf[9], red);
}

// =====================================================================
// FALLBACK path (small workspace): recompute tp from inputs each pass.
// =====================================================================
__device__ __forceinline__ void acc2(float l0, float l1, float t, float e,
                                     u32 d1, u32* hist) {
  if (t == 1.0f) {
    float p  = sig_p(l0, l1);
    u32 bits = __float_as_uint(p * (1.0f + e));
    if ((bits >> 21) == d1) atomicAdd(&hist[(bits >> 10) & 0x7FFu], 1u);
  }
}

__global__ __launch_bounds__(TPB) void k_pass2(const float* __restrict__ logits,
                                               const float* __restrict__ target,
                                               const float* __restrict__ eps,
                                               u32* __restrict__ ws) {
  __shared__ u32 hist[BINS_A];
  int b = blockIdx.x / CPB, chunk = blockIdx.x % CPB;
  u32* acc = ws + OFF_ACC + (size_t)b * ACC_ST;
  u32 d1 = acc[2];
  if (d1 == FG_SENTINEL) return;
  for (int i = threadIdx.x; i < BINS_A; i += TPB) hist[i] = 0u;
  __syncthreads();
  const float* l0p = logits + (size_t)b * NPIX2;
  const float* l1p = l0p + NPIX;
  const float* tpp = target + (size_t)b * NPIX;
  const float* epp = eps    + (size_t)b * NPIX;
  int base = chunk * EPB;
  #pragma unroll
  for (int it = 0; it < PITER; ++it) {
    int i = base + it * (TPB * 4) + threadIdx.x * 4;
    float4 a0 = *(const float4*)(l0p + i);
    float4 a1 = *(const float4*)(l1p + i);
    float4 tt = *(const float4*)(tpp + i);
    float4 ee = *(const float4*)(epp + i);
    acc2(a0.x, a1.x, tt.x, ee.x, d1, hist);
    acc2(a0.y, a1.y, tt.y, ee.y, d1, hist);
    acc2(a0.z, a1.z, tt.z, ee.z, d1, hist);
    acc2(a0.w, a1.w, tt.w, ee.w, d1, hist);
  }
  __syncthreads();
  u32* gh = ws + OFF_H2 + (size_t)b * BINS_A;
  for (int i = threadIdx.x; i < BINS_A; i += TPB) {
    u32 c = hist[i];
    if (c) atomicAdd(&gh[i], c);
  }
}

__device__ __forceinline__ void acc3(float l0, float l1, float t, float e,
                                     u32 key, u32* hist) {
  if (t == 1.0f) {
    float p  = sig_p(l0, l1);
    u32 bits = __float_as_uint(p * (1.0f + e));
    if ((bits >> 10) == key) atomicAdd(&hist[bits & 0x3FFu], 1u);
  }
}

__global__ __launch_bounds__(TPB) void k_pass3(const float* __restrict__ logits,
                                               const float* __restrict__ target,
                                               const float* __restrict__ eps,
                                               u32* __restrict__ ws) {
  __shared__ u32 hist[BINS_C];
  int b = blockIdx.x / CPB, chunk = blockIdx.x % CPB;
  u32* acc = ws + OFF_ACC + (size_t)b * ACC_ST;
  u32 d1 = acc[2];
  if (d1 == FG_SENTINEL) return;
  u32 key = (d1 << 11) | acc[4];
  for (int i = threadIdx.x; i < BINS_C; i += TPB) hist[i] = 0u;
  __syncthreads();
  const float* l0p = logits + (size_t)b * NPIX2;
  const float* l1p = l0p + NPIX;
  const float* tpp = target + (size_t)b * NPIX;
  const float* epp = eps    + (size_t)b * NPIX;
  int base = chunk * EPB;
  #pragma unroll
  for (int it = 0; it < PITER; ++it) {
    int i = base + it * (TPB * 4) + threadIdx.x * 4;
    float4 a0 = *(const float4*)(l0p + i);
    float4 a1 = *(const float4*)(l1p + i);
    float4 tt = *(const float4*)(tpp + i);
    float4 ee = *(const float4*)(epp + i);
    acc3(a0.x, a1.x, tt.x, ee.x, key, hist);
    acc3(a0.y, a1.y, tt.y, ee.y, key, hist);
    acc3(a0.z, a1.z, tt.z, ee.z, key, hist);
    acc3(a0.w, a1.w, tt.w, ee.w, key, hist);
  }
  __syncthreads();
  u32* gh = ws + OFF_H3 + (size_t)b * BINS_C;
  for (int i = threadIdx.x; i < BINS_C; i += TPB) {
    u32 c = hist[i];
    if (c) atomicAdd(&gh[i], c);
  }
}

__device__ __forceinline__ void acc4(float l0, float l1, float t, float e, float thr,
                                     float& inter, float& cnt, float& p2ig) {
  if (t == 1.0f) {
    float p  = sig_p(l0, l1);
    float tp = p * (1.0f + e);
    if (tp <= thr) { inter += p; cnt += 1.0f; }
    else           { p2ig += p * p; }
  }
}

__global__ __launch_bounds__(TPB) void k_final(const float* __restrict__ logits,
                                               const float* __restrict__ target,
                                               const float* __restrict__ eps,
                                               u32* __restrict__ ws) {
  __shared__ float red[NWAVES];
  int b = blockIdx.x / CPB, chunk = blockIdx.x % CPB;
  u32* acc = ws + OFF_ACC + (size_t)b * ACC_ST;
  float thr = __uint_as_float(acc[6]);
  const float* l0p = logits + (size_t)b * NPIX2;
  const float* l1p = l0p + NPIX;
  const float* tpp = target + (size_t)b * NPIX;
  const float* epp = eps    + (size_t)b * NPIX;
  int base = chunk * EPB;
  float inter = 0.0f, cnt = 0.0f, p2ig = 0.0f;
  #pragma unroll
  for (int it = 0; it < PITER; ++it) {
    int i = base + it * (TPB * 4) + threadIdx.x * 4;
    float4 a0 = *(const float4*)(l0p + i);
    float4 a1 = *(const float4*)(l1p + i);
    float4 tt = *(const float4*)(tpp + i);
    float4 ee = *(const float4*)(epp + i);
    acc4(a0.x, a1.x, tt.x, ee.x, thr, inter, cnt, p2ig);
    acc4(a0.y, a1.y, tt.y, ee.y, thr, inter, cnt, p2ig);
    acc4(a0.z, a1.z, tt.z, ee.z, thr, inter, cnt, p2ig);
    acc4(a0.w, a1.w, tt.w, ee.w, thr, inter, cnt, p2ig);
  }
  float* accf = (float*)acc;
  block_atomic_add(inter, &accf[7], red);
  block_atomic_add(p2ig,  &accf[8], red);
  block_atomic_add(cnt,   &accf[9], red);
}

// ---------------- selection over histograms (one block per batch) ----------------
__global__ __launch_bounds__(256) void k_select(u32* __restrict__ ws, int mode) {
  __shared__ u32 part[256];
  __shared__ u32 pref[256];
  int b = blockIdx.x;
  u32* acc = ws + OFF_ACC + (size_t)b * ACC_ST;
  float* accf = (float*)acc;
  int tid = threadIdx.x;

  u32 k = 0, d1 = 0, d2 = 0;
  const u32* h = nullptr;
  int bins = BINS_A;
  if (mode == 0) {
    float nfg = accf[1];
    if (nfg == 0.0f) {                        // no foreground: thr = +inf
      if (tid == 0) { acc[2] = FG_SENTINEL; acc[6] = 0x7F800000u; }
      return;
    }
    int k0 = (int)(nfg * 0.5f);               // matches int32 truncation in reference
    if (k0 < 1) k0 = 1;
    k = (u32)k0;
    h = ws + OFF_H1 + (size_t)b * BINS_A;
  } else {
    d1 = acc[2];
    if (d1 == FG_SENTINEL) return;
    if (mode == 1) { k = acc[3]; h = ws + OFF_H2 + (size_t)b * BINS_A; }
    else { d2 = acc[4]; k = acc[5]; h = ws + OFF_H3 + (size_t)b * BINS_C; bins = BINS_C; }
  }

  int per = bins / 256;                       // 8 or 4 bins per thread
  u32 s = 0;
  for (int j = 0; j < per; ++j) s += h[tid * per + j];
  part[tid] = s;
  __syncthreads();
  if (tid == 0) {
    u32 run = 0;
    for (int i = 0; i < 256; ++i) { pref[i] = run; run += part[i]; }
  }
  __syncthreads();
  u32 p0 = pref[tid];
  if (k > p0 && k <= p0 + part[tid]) {        // exactly one owner thread
    u32 run = p0;
    for (int j = 0; j < per; ++j) {
      u32 c = h[tid * per + j];
      if (k <= run + c) {
        u32 d = (u32)(tid * per + j);
        u32 r = k - run;                      // 1-based rank within bin
        if      (mode == 0) { acc[2] = d; acc[3] = r; }
        else if (mode == 1) { acc[4] = d; acc[5] = r; }
        else                { acc[6] = (d1 << 21) | (d2 << 10) | d; }
        break;
      }
      run += c;
    }
  }
}

// ---------------- dice + mean ----------------
__global__ void k_out(const u32* __restrict__ ws, float* __restrict__ out) {
  int b = threadIdx.x;                        // 32 threads, one wave
  const float* accf = (const float*)(ws + OFF_ACC + (size_t)b * ACC_ST);
  float sumP2 = accf[0];
  float inter = accf[7];
  float p2ig  = accf[8];
  float cnt   = accf[9];
  float uni  = (sumP2 - p2ig) + cnt;          // sum(probs_k^2) + sum(target_k^2)
  float dice = (2.0f * inter + SMOOTHF) / (uni + SMOOTHF);
  #pragma unroll
  for (int off = 16; off >= 1; off >>= 1) dice += __shfl_xor(dice, off, 32);
  if (b == 0) out[0] = 1.0f - dice * (1.0f / 32.0f);
}

extern "C" void kernel_launch(void* const* d_in, const int* in_sizes, int n_in,
                              void* d_out, int out_size, void* d_ws, size_t ws_size,
                              hipStream_t stream) {
  (void)in_sizes; (void)n_in; (void)out_size;
  const float* logits = (const float*)d_in[0];
  const float* target = (const float*)d_in[1];
  const float* eps    = (const float*)d_in[2];
  u32*   ws  = (u32*)d_ws;
  float* out = (float*)d_out;

  const bool cached = ws_size >= (size_t)WS_U32_CACHED * sizeof(u32);

  k_zero<<<(WS_U32 + 255) / 256, 256, 0, stream>>>(ws);
  if (cached) {
    k_pass1<true><<<NBATCH * CPB, TPB, 0, stream>>>(logits, target, eps, ws);
    k_select<<<NBATCH, 256, 0, stream>>>(ws, 0);
    k_pass2c<<<NBATCH * CPB, TPB, 0, stream>>>(ws);
    k_select<<<NBATCH, 256, 0, stream>>>(ws, 1);
    k_pass3c<<<NBATCH * CPB, TPB, 0, stream>>>(ws);
    k_select<<<NBATCH, 256, 0, stream>>>(ws, 2);
    k_finalc<<<NBATCH * CPB, TPB, 0, stream>>>(eps, ws);
  } else {
    k_pass1<false><<<NBATCH * CPB, TPB, 0, stream>>>(logits, target, eps, ws);
    k_select<<<NBATCH, 256, 0, stream>>>(ws, 0);
    k_pass2<<<NBATCH * CPB, TPB, 0, stream>>>(logits, target, eps, ws);
    k_select<<<NBATCH, 256, 0, stream>>>(ws, 1);
    k_pass3<<<NBATCH * CPB, TPB, 0, stream>>>(logits, target, eps, ws);
    k_select<<<NBATCH, 256, 0, stream>>>(ws, 2);
    k_final<<<NBATCH * CPB, TPB, 0, stream>>>(logits, target, eps, ws);
  }
  k_out<<<1, 32, 0, stream>>>(ws, out);
}